// LieSpline_42674795053188
// MI455X (gfx1250) — compile-verified
//
#include <hip/hip_runtime.h>

#define LIE_EPS  1e-6f
#define LIE_EPS2 1e-12f

struct V3 { float x, y, z; };
struct Q4 { float x, y, z, w; };
struct SE3 { V3 t; Q4 q; };
struct Xi  { V3 tau; V3 phi; };

__device__ __forceinline__ V3 vadd(V3 a, V3 b) { return {a.x + b.x, a.y + b.y, a.z + b.z}; }
__device__ __forceinline__ V3 vscale(V3 a, float s) { return {a.x * s, a.y * s, a.z * s}; }
__device__ __forceinline__ float vdot(V3 a, V3 b) { return a.x * b.x + a.y * b.y + a.z * b.z; }
__device__ __forceinline__ V3 vcross(V3 a, V3 b) {
    return { a.y * b.z - a.z * b.y, a.z * b.x - a.x * b.z, a.x * b.y - a.y * b.x };
}

__device__ __forceinline__ Q4 qmul(Q4 a, Q4 b) {
    return { a.w * b.x + a.x * b.w + a.y * b.z - a.z * b.y,
             a.w * b.y + a.y * b.w + a.z * b.x - a.x * b.z,
             a.w * b.z + a.z * b.w + a.x * b.y - a.y * b.x,
             a.w * b.w - a.x * b.x - a.y * b.y - a.z * b.z };
}

__device__ __forceinline__ V3 qrot(Q4 q, V3 v) {
    V3 qv = { q.x, q.y, q.z };
    V3 t  = vscale(vcross(qv, v), 2.0f);
    return vadd(vadd(v, vscale(t, q.w)), vcross(qv, t));
}

// ---------- wave-uniform path (amortized 32x, rides SALU/libm precision) ----------

__device__ __forceinline__ V3 so3_log(Q4 q) {
    V3 qv = { q.x, q.y, q.z };
    float n2 = vdot(qv, qv);
    bool small = n2 < LIE_EPS2;
    float n = sqrtf(small ? 1.0f : n2);
    float factor = small
        ? (2.0f / q.w - (2.0f / 3.0f) * n2 / (q.w * q.w * q.w))
        : (2.0f * atan2f(n, q.w) / n);
    return vscale(qv, factor);
}

__device__ __forceinline__ V3 jl_inv(V3 phi, V3 v) {
    float t2 = vdot(phi, phi);
    bool small = t2 < LIE_EPS2;
    float t2s = small ? 1.0f : t2;
    float theta = sqrtf(t2s);
    float sn = sinf(theta);
    float s_safe = (fabsf(sn) < LIE_EPS) ? LIE_EPS : sn;
    float c = small ? (1.0f / 12.0f + t2 * (1.0f / 720.0f))
                    : (1.0f / t2s - (1.0f + cosf(theta)) / (2.0f * theta * s_safe));
    V3 c1 = vcross(phi, v);
    return vadd(vadd(v, vscale(c1, -0.5f)), vscale(vcross(phi, c1), c));
}

__device__ __forceinline__ SE3 se3_inv(SE3 T) {
    Q4 qi = { -T.q.x, -T.q.y, -T.q.z, T.q.w };
    V3 ti = qrot(qi, T.t);
    return { { -ti.x, -ti.y, -ti.z }, qi };
}

__device__ __forceinline__ SE3 se3_mul(SE3 A, SE3 B) {
    return { vadd(A.t, qrot(A.q, B.t)), qmul(A.q, B.q) };
}

__device__ __forceinline__ Xi se3_log(SE3 T) {
    V3 phi = so3_log(T.q);
    return { jl_inv(phi, T.t), phi };
}

// ---------- per-lane path: |theta| <= 2*pi guaranteed -> hardware v_sin/v_cos ----------

__device__ __forceinline__ Q4 so3_exp_fast(V3 phi) {
    float t2 = vdot(phi, phi);
    bool small = t2 < LIE_EPS2;
    float theta = sqrtf(small ? 1.0f : t2);
    float half = 0.5f * theta;                    // in [0, pi]
    float s = small ? (0.5f - t2 * (1.0f / 48.0f)) : (__sinf(half) / theta);
    float w = small ? (1.0f - t2 * (1.0f / 8.0f)) : __cosf(half);
    return { s * phi.x, s * phi.y, s * phi.z, w };
}

__device__ __forceinline__ V3 jl_fast(V3 phi, V3 v) {
    float t2 = vdot(phi, phi);
    bool small = t2 < LIE_EPS2;
    float t2s = small ? 1.0f : t2;
    float theta = sqrtf(t2s);                     // in [0, 2*pi]
    float a = small ? (0.5f - t2 * (1.0f / 24.0f)) : ((1.0f - __cosf(theta)) / t2s);
    float b = small ? (1.0f / 6.0f - t2 * (1.0f / 120.0f))
                    : ((theta - __sinf(theta)) / (t2s * theta));
    V3 c1 = vcross(phi, v);
    return vadd(vadd(v, vscale(c1, a)), vscale(vcross(phi, c1), b));
}

__device__ __forceinline__ SE3 se3_exp_fast(Xi x) {
    Q4 q = so3_exp_fast(x.phi);
    V3 t = jl_fast(x.phi, x.tau);
    return { t, q };
}

__device__ __forceinline__ SE3 load_pose(const float* __restrict__ p, int idx) {
    const float* s = p + (long long)idx * 7;
    SE3 T;
    T.t = { s[0], s[1], s[2] };
    T.q = { s[3], s[4], s[5], s[6] };
    return T;
}

typedef __attribute__((address_space(3))) float lds_float_t;

__global__ __launch_bounds__(256) void LieSpline_kernel(
    const float* __restrict__ poses,   // B x N x 7
    const float* __restrict__ time,    // K
    float* __restrict__ out,           // B x S x K x 7
    int B, int N, int K)
{
    const int S = N - 1;
    __shared__ float s_time[32];

    // --- CDNA5 async-to-LDS staging of the K time samples (ASYNCcnt path) ---
    if ((int)threadIdx.x < K) {
        unsigned lds_off;
        {
            lds_float_t* p3 = (lds_float_t*)&s_time[threadIdx.x];
            lds_off = (unsigned)(unsigned long long)p3;
        }
        const float* g = time + threadIdx.x;
        asm volatile("global_load_async_to_lds_b32 %0, %1, off"
                     :: "v"(lds_off), "v"(g)
                     : "memory");
    }
    asm volatile("s_wait_asynccnt 0" ::: "memory");
    __syncthreads();

    // wave32 mapping: lane = k sample; (b,s) is wave-uniform -> scalar loads
    const int lane = threadIdx.x & 31;
    int wave = (int)(blockIdx.x * (blockDim.x >> 5)) + (int)(threadIdx.x >> 5);
    wave = __builtin_amdgcn_readfirstlane(wave);
    if (wave >= B * S) return;
    const int b = wave / S;
    const int s = wave - b * S;

    // padded pose index: padded[j] maps to input clamp(j-1, 0, N-1)
    const int base = b * N;
    int i0 = s - 1; i0 = i0 < 0 ? 0 : i0;                    // j = s
    int i1 = s;                                              // j = s+1
    int i2 = s + 1;                                          // j = s+2
    int i3 = s + 2; i3 = i3 > N - 1 ? N - 1 : i3;            // j = s+3

    SE3 P0 = load_pose(poses, base + i0);
    SE3 P1 = load_pose(poses, base + i1);
    SE3 P2 = load_pose(poses, base + i2);
    SE3 P3 = load_pose(poses, base + i3);

    // wave-uniform: executed once per wave, amortized over 32 samples
    Xi d0 = se3_log(se3_mul(se3_inv(P0), P1));
    Xi d1 = se3_log(se3_mul(se3_inv(P1), P2));
    Xi d2 = se3_log(se3_mul(se3_inv(P2), P3));

    if (lane < K) {
        float u  = s_time[lane];
        float u2 = u * u;
        float u3 = u2 * u;
        // w = (1/6) * M @ [1, u, u^2, u^3]
        float w0 = (5.0f + 3.0f * u - 3.0f * u2 + u3)        * (1.0f / 6.0f);
        float w1 = (1.0f + 3.0f * u + 3.0f * u2 - 2.0f * u3) * (1.0f / 6.0f);
        float w2 = u3 * (1.0f / 6.0f);

        SE3 A0 = se3_exp_fast(Xi{ vscale(d0.tau, w0), vscale(d0.phi, w0) });
        SE3 A1 = se3_exp_fast(Xi{ vscale(d1.tau, w1), vscale(d1.phi, w1) });
        SE3 A2 = se3_exp_fast(Xi{ vscale(d2.tau, w2), vscale(d2.phi, w2) });

        SE3 T = se3_mul(se3_mul(se3_mul(P0, A0), A1), A2);

        // out[b, s, k, :], one contiguous 896B span per wave
        long long o = (((long long)b * S + s) * K + lane) * 7;
        out[o + 0] = T.t.x;
        out[o + 1] = T.t.y;
        out[o + 2] = T.t.z;
        out[o + 3] = T.q.x;
        out[o + 4] = T.q.y;
        out[o + 5] = T.q.z;
        out[o + 6] = T.q.w;
    }
}

extern "C" void kernel_launch(void* const* d_in, const int* in_sizes, int n_in,
                              void* d_out, int out_size, void* d_ws, size_t ws_size,
                              hipStream_t stream) {
    const float* poses = (const float*)d_in[0];  // (B, N, 7) f32
    const float* time  = (const float*)d_in[1];  // (1, 1, K) f32
    float* out = (float*)d_out;                  // (B, S*K, 7) f32

    const int B = 32;
    int K = in_sizes[1];                 // 32
    int N = in_sizes[0] / (B * 7);       // 2048
    int S = N - 1;                       // 2047

    long long waves = (long long)B * S;               // 65504
    const int wavesPerBlock = 8;                      // 256 threads
    int blocks = (int)((waves + wavesPerBlock - 1) / wavesPerBlock);

    LieSpline_kernel<<<blocks, wavesPerBlock * 32, 0, stream>>>(
        poses, time, out, B, N, K);
}